// SceneEncoder_70875550318620
// MI455X (gfx1250) — compile-verified
//
#include <hip/hip_runtime.h>
#include <hip/hip_bf16.h>
#include <math.h>

// ---------------------------------------------------------------------------
// SceneEncoder for MI455X (gfx1250, wave32, WMMA + TDM).
// Algebraic reduction: pairwise MLP input  W1@[x_i; x_i - x_j] + b1
//   == A[b,i] - C[b,j] with A = x@(W1a+W1b)^T + b1, C = x@W1b^T.
// agg = W2 @ (sum_j attn * gelu(A - C)) + b2  (since sum_j attn == 1).
// Dense GEMMs: V_WMMA_F32_16X16X4_F32; A-tile staged to LDS via TDM
// (tensor_load_to_lds, 1-D contiguous descriptor, 6-arg clang-23 builtin).
// ---------------------------------------------------------------------------

typedef float v2f __attribute__((ext_vector_type(2)));
typedef float v8f __attribute__((ext_vector_type(8)));
typedef unsigned int v4u __attribute__((ext_vector_type(4)));
typedef int v8i __attribute__((ext_vector_type(8)));
typedef int v4i __attribute__((ext_vector_type(4)));

#define DD   256
#define KK   64
#define BB   32
#define MM   2048   // B*K rows
#define LSZ  128

#if __has_builtin(__builtin_amdgcn_tensor_load_to_lds) && \
    __has_builtin(__builtin_amdgcn_s_wait_tensorcnt)
#define USE_TDM 1
#else
#define USE_TDM 0
#endif

__device__ __forceinline__ float gelu_exact(float x) {
    // jax.nn.gelu(approximate=False) = 0.5*x*(1+erf(x/sqrt(2)))
    return 0.5f * x * (1.0f + erff(x * 0.70710678118654752f));
}
__device__ __forceinline__ float softplus_stable(float x) {
    // log(1+exp(x)) stable
    return fmaxf(x, 0.0f) + log1pf(expf(-fabsf(x)));
}

// ---------------------------------------------------------------------------
// WMMA fp32 GEMM:  Out[m,n] = act((sum_k X[m,k]*W[n,k] + bias[n])*scale)
//                              + (resid ? resid[m,n] : 0)
// Block = 128 threads = 4 waves; block owns 16 rows x 64 cols; each wave one
// 16x16 tile. The 16xKd A-tile (contiguous in memory) is staged into LDS once
// per block — via the Tensor Data Mover when available — and shared by all
// 4 waves; A-fragments then come from LDS (ds loads), B from global (L2-hot).
// Fragment layout per CDNA5 ISA 7.12.2 (fp32 16x16x4):
//   A: lane(0..15)=row M, VGPR0 = K=(hi?2:0), VGPR1 = K=(hi?3:1)
//   B: lane(0..15)=col N, same K split (B[k,n] = W[n,k])
//   C/D: VGPR r -> row r + 8*hi, col = lane&15
// ---------------------------------------------------------------------------
__global__ void wmma_gemm_kernel(const float* __restrict__ Xm,
                                 const float* __restrict__ Wm, int ldw,
                                 const float* __restrict__ bias,
                                 const float* __restrict__ resid,
                                 float* __restrict__ Out,
                                 int M, int N, int Kd, int act, float scale)
{
    __shared__ float xtile[16 * DD];      // 16 rows x Kd (Kd <= 256), 16 KB

    const int wave = threadIdx.x >> 5;
    const int lane = threadIdx.x & 31;
    const int idx  = lane & 15;
    const int hi   = lane >> 4;           // 0: lanes 0-15, 1: lanes 16-31
    const int m0   = blockIdx.x * 16;
    const int n0   = blockIdx.y * 64 + wave * 16;
    const int tile_elems = 16 * Kd;       // contiguous block of X

#if USE_TDM
    if (threadIdx.x < 32) {
        // 1-D TDM descriptor: copy 16*Kd contiguous f32 from global to LDS.
        unsigned long long ga =
            (unsigned long long)(const void*)(Xm + (size_t)m0 * Kd);
        unsigned int ldsa = (unsigned int)(__SIZE_TYPE__)&xtile[0];
        unsigned int nel  = (unsigned int)tile_elems;      // <= 4096
        v4u g0;
        g0.x = 1u;                                          // count=1, flags=0
        g0.y = ldsa;                                        // lds_addr [63:32]
        g0.z = (unsigned int)(ga & 0xFFFFFFFFu);            // global_addr lo
        g0.w = (unsigned int)((ga >> 32) & 0x01FFFFFFu)     // global_addr hi
             | 0x80000000u;                                 // type=2 [127:126]
        v8i g1;
        g1[0] = (int)(2u << 16);                            // data_size=4B
        g1[1] = (int)((nel & 0xFFFFu) << 16);               // tensor_dim0 lo16
        g1[2] = (int)(((nel >> 16) & 0xFFFFu) | (1u << 16));// t_dim0 hi,t_dim1=1
        g1[3] = (int)((nel & 0xFFFFu) << 16);               // tile_dim0
        g1[4] = 0;                                          // tile_dim1/2 = 0
        g1[5] = (int)nel;                                   // dim0_stride lo32
        g1[6] = 0;
        g1[7] = 0;
        v4i gz4 = {0, 0, 0, 0};
        v8i gz8 = {0, 0, 0, 0, 0, 0, 0, 0};
        __builtin_amdgcn_tensor_load_to_lds(g0, g1, gz4, gz4, gz8, 0);
        __builtin_amdgcn_s_wait_tensorcnt(0);
    }
#else
    for (int i = threadIdx.x; i < tile_elems; i += blockDim.x)
        xtile[i] = Xm[(size_t)m0 * Kd + i];     // coalesced staging
#endif
    __syncthreads();

    const float* arow = &xtile[idx * Kd];
    const float* wrow = Wm + (size_t)(n0 + idx) * ldw;
    const int kb = 2 * hi;

    v8f acc = {0.f, 0.f, 0.f, 0.f, 0.f, 0.f, 0.f, 0.f};
    for (int k0 = 0; k0 < Kd; k0 += 4) {
        v2f a, b;
        a.x = arow[k0 + kb];                    // ds_load from staged tile
        a.y = arow[k0 + kb + 1];
        b.x = wrow[k0 + kb];
        b.y = wrow[k0 + kb + 1];
        acc = __builtin_amdgcn_wmma_f32_16x16x4_f32(
            /*neg_a=*/false, a, /*neg_b=*/false, b,
            /*c_mod=*/(short)0, acc, /*reuse_a=*/false, /*reuse_b=*/false);
    }

    const float bn = bias ? bias[n0 + idx] : 0.0f;
#pragma unroll
    for (int r = 0; r < 8; ++r) {
        const int row = m0 + r + 8 * hi;
        float v = (acc[r] + bn) * scale;
        if (act == 1)      v = gelu_exact(v);
        else if (act == 2) v = softplus_stable(v);
        const size_t o = (size_t)row * N + (n0 + idx);
        if (resid) v += resid[o];
        Out[o] = v;
    }
}

// Wsum[d,e] = rv_w1[d, e] + rv_w1[d, D+e]   (combine xi-half and r-half)
__global__ void wsum_kernel(const float* __restrict__ rv1, float* __restrict__ wsum)
{
    const int i = blockIdx.x * blockDim.x + threadIdx.x;   // D*D threads
    const int d = i >> 8, e = i & 255;
    wsum[i] = rv1[d * 512 + e] + rv1[d * 512 + 256 + e];
}

// Per (b,i): softmax over j of q.k, then g[d] = sum_j p_j * gelu(A[i,d]-C[j,d])
// Block = 256 threads (8 waves), one block per row bi = b*K+i.
__global__ void attn_pair_kernel(const float* __restrict__ qb,
                                 const float* __restrict__ kb,
                                 const float* __restrict__ Ab,
                                 const float* __restrict__ Cb,
                                 float* __restrict__ gb)
{
    __shared__ float qs[DD];
    __shared__ float sc[KK];
    __shared__ float red[2];

    const int bi = blockIdx.x;           // b*K + i
    const int b  = bi >> 6;              // / K
    const int t  = threadIdx.x;

    qs[t] = qb[(size_t)bi * DD + t];     // q already has D^-0.5 folded in
    __syncthreads();

    if (t < KK) {
        const float* krow = kb + (size_t)(b * KK + t) * DD;
        float dot = 0.0f;
        for (int d = 0; d < DD; ++d) dot = fmaf(qs[d], krow[d], dot);
        sc[t] = dot;
    }
    __syncthreads();
    if (t == 0) {
        float m = sc[0];
        for (int j = 1; j < KK; ++j) m = fmaxf(m, sc[j]);
        red[0] = m;
    }
    __syncthreads();
    if (t < KK) sc[t] = expf(sc[t] - red[0]);
    __syncthreads();
    if (t == 0) {
        float s = 0.0f;
        for (int j = 0; j < KK; ++j) s += sc[j];
        red[1] = 1.0f / s;
    }
    __syncthreads();

    const float inv = red[1];
    const float a   = Ab[(size_t)bi * DD + t];
    const float* cc = Cb + (size_t)(b * KK) * DD + t;   // coalesced over t
    float acc = 0.0f;
    for (int j = 0; j < KK; ++j) {
        const float u = a - cc[(size_t)j * DD];
        acc = fmaf(sc[j] * inv, gelu_exact(u), acc);
    }
    gb[(size_t)bi * DD + t] = acc;
}

// LayerNorm over D=256: h = (v-mean)/sqrt(var+1e-5)*g + b. Block per row.
__global__ void ln_kernel(const float* __restrict__ xr,
                          const float* __restrict__ gam,
                          const float* __restrict__ bet,
                          float* __restrict__ h)
{
    __shared__ float red[DD];
    __shared__ float mv[2];
    const int row = blockIdx.x;
    const int t   = threadIdx.x;
    const float v = xr[(size_t)row * DD + t];

    red[t] = v;
    __syncthreads();
    for (int s = 128; s > 0; s >>= 1) { if (t < s) red[t] += red[t + s]; __syncthreads(); }
    if (t == 0) mv[0] = red[0] * (1.0f / DD);
    __syncthreads();

    const float d = v - mv[0];
    red[t] = d * d;
    __syncthreads();
    for (int s = 128; s > 0; s >>= 1) { if (t < s) red[t] += red[t + s]; __syncthreads(); }
    if (t == 0) mv[1] = red[0] * (1.0f / DD);
    __syncthreads();

    h[(size_t)row * DD + t] = d * rsqrtf(mv[1] + 1e-5f) * gam[t] + bet[t];
}

// s[b,d] = sum_i x[b,i,d]  (token sum). Block per b, 256 threads.
__global__ void sumk_kernel(const float* __restrict__ x, float* __restrict__ s)
{
    const int b = blockIdx.x, t = threadIdx.x;
    float acc = 0.0f;
    for (int i = 0; i < KK; ++i) acc += x[(size_t)(b * KK + i) * DD + t];
    s[(size_t)b * DD + t] = acc;
}

extern "C" void kernel_launch(void* const* d_in, const int* in_sizes, int n_in,
                              void* d_out, int out_size, void* d_ws, size_t ws_size,
                              hipStream_t stream)
{
    (void)in_sizes; (void)n_in; (void)out_size; (void)ws_size;

    const float* X      = (const float*)d_in[0];
    const float* k_w    = (const float*)d_in[1];
    const float* q_w    = (const float*)d_in[2];
    const float* rv_w1  = (const float*)d_in[3];
    const float* rv_b1  = (const float*)d_in[4];
    const float* rv_w2  = (const float*)d_in[5];
    const float* rv_b2  = (const float*)d_in[6];
    const float* ln_g   = (const float*)d_in[7];
    const float* ln_b   = (const float*)d_in[8];
    const float* mlp_w1 = (const float*)d_in[9];
    const float* mlp_b1 = (const float*)d_in[10];
    const float* mlp_w2 = (const float*)d_in[11];
    const float* mlp_b2 = (const float*)d_in[12];
    const float* proj_w = (const float*)d_in[13];
    const float* proj_b = (const float*)d_in[14];
    const float* mu_w   = (const float*)d_in[15];
    const float* mu_b   = (const float*)d_in[16];
    const float* sp_w   = (const float*)d_in[17];
    const float* sp_b   = (const float*)d_in[18];
    float* out = (float*)d_out;

    const size_t RC = (size_t)MM * DD;      // 524288 floats per activation buf
    float* ws   = (float*)d_ws;
    float* xA   = ws;
    float* xB   = xA + RC;
    float* xr   = xB + RC;
    float* hbuf = xr + RC;
    float* Ab   = hbuf + RC;
    float* Cb   = Ab + RC;
    float* qb   = Cb + RC;
    float* kbuf = qb + RC;
    float* gb   = kbuf + RC;
    float* wsum = gb + RC;                  // D*D = 65536
    float* sbuf = wsum + (size_t)DD * DD;   // B*D = 8192
    float* hh   = sbuf + (size_t)BB * DD;   // B*LS = 4096
    float* t1   = Ab;                       // reuse A for MLP hidden

    const float qscale = 0.0625f;           // D^-0.5 = 1/16
    const int NL_ = 2;

    auto gemm = [&](const float* Xm, const float* Wm, int ldw, const float* bias,
                    const float* resid, float* Out, int Mq, int Nq, int Kd,
                    int act, float sc) {
        dim3 grid(Mq / 16, Nq / 64);
        wmma_gemm_kernel<<<grid, dim3(128), 0, stream>>>(
            Xm, Wm, ldw, bias, resid, Out, Mq, Nq, Kd, act, sc);
    };

    const float* xin = X;
    float* xout = xA;
    for (int l = 0; l < NL_; ++l) {
        const float* rv1 = rv_w1 + (size_t)l * DD * 2 * DD;
        // Wsum = W1a + W1b
        wsum_kernel<<<(DD * DD) / 256, 256, 0, stream>>>(rv1, wsum);
        // A = x @ Wsum^T + b1 ; C = x @ W1b^T (W1b = strided right half of rv_w1)
        gemm(xin, wsum,              DD,     rv_b1 + (size_t)l * DD, nullptr, Ab,   MM, DD, DD, 0, 1.0f);
        gemm(xin, rv1 + DD,          2 * DD, nullptr,                nullptr, Cb,   MM, DD, DD, 0, 1.0f);
        // q (scaled), k
        gemm(xin, q_w + (size_t)l * DD * DD, DD, nullptr,            nullptr, qb,   MM, DD, DD, 0, qscale);
        gemm(xin, k_w + (size_t)l * DD * DD, DD, nullptr,            nullptr, kbuf, MM, DD, DD, 0, 1.0f);
        // softmax + pairwise gelu-weighted reduction -> g
        attn_pair_kernel<<<MM, DD, 0, stream>>>(qb, kbuf, Ab, Cb, gb);
        // xr = x + (g @ W2^T + b2)
        gemm(gb, rv_w2 + (size_t)l * DD * DD, DD, rv_b2 + (size_t)l * DD, xin, xr, MM, DD, DD, 0, 1.0f);
        // h = LN(xr)
        ln_kernel<<<MM, DD, 0, stream>>>(xr, ln_g + (size_t)l * DD, ln_b + (size_t)l * DD, hbuf);
        // MLP + residual: xout = xr + (gelu(h@W1^T+b1) @ W2^T + b2)
        gemm(hbuf, mlp_w1 + (size_t)l * DD * DD, DD, mlp_b1 + (size_t)l * DD, nullptr, t1,   MM, DD, DD, 1, 1.0f);
        gemm(t1,   mlp_w2 + (size_t)l * DD * DD, DD, mlp_b2 + (size_t)l * DD, xr,      xout, MM, DD, DD, 0, 1.0f);
        xin  = xout;
        xout = (xout == xA) ? xB : xA;
    }

    // Head: s = sum over tokens; h = gelu(s@proj^T+b); mu, softplus to d_out.
    sumk_kernel<<<BB, DD, 0, stream>>>(xin, sbuf);
    gemm(sbuf, proj_w, DD,  proj_b, nullptr, hh,             BB, LSZ, DD,  1, 1.0f);
    gemm(hh,   mu_w,   LSZ, mu_b,   nullptr, out,            BB, LSZ, LSZ, 0, 1.0f);
    gemm(hh,   sp_w,   LSZ, sp_b,   nullptr, out + BB * LSZ, BB, LSZ, LSZ, 2, 1.0f);
}